// SymmetryDetection_4501125726670
// MI455X (gfx1250) — compile-verified
//
#include <hip/hip_runtime.h>
#include <hip/hip_bf16.h>
#include <math.h>

#define NANGLE 24
#define DSC_POLAR 4
#define SYM_POLAR 3
#define HW 256
#define CC 512
#define EPSV 1e-12f
#define PI_F 3.14159265358979323846f

typedef __attribute__((ext_vector_type(2))) float v2f;
typedef __attribute__((ext_vector_type(8))) float v8f;

// ---------------------------------------------------------------------------
// Kernel A: Gram matrix G = F^T F  (F is 512 x 256, G is 256 x 256)
// One wave per 16x16 output tile, K-loop of v_wmma_f32_16x16x4_f32.
// A-layout (16x4 f32): lane 0-15 -> M=lane, VGPR0/1 = K0/K1; lanes 16-31 = K2/K3.
// B-layout mirrors with N=lane. C/D: VGPR v -> M=v (+8 for lanes>=16), N=lane&15.
// ---------------------------------------------------------------------------
__global__ __launch_bounds__(256) void gram_kernel(const float* __restrict__ F,
                                                   float* __restrict__ G) {
  const int lane = threadIdx.x & 31;
  const int wave = threadIdx.x >> 5;
  const int tile = blockIdx.x * 8 + wave;   // 32 blocks * 8 waves = 256 tiles
  const int I = tile >> 4;                  // row tile
  const int J = tile & 15;                  // col tile
  const int colA = I * 16 + (lane & 15);
  const int colB = J * 16 + (lane & 15);
  const int khalf = (lane >> 4) << 1;       // 0 for lanes 0-15, 2 for lanes 16-31

  v8f acc = {};
  for (int k0 = 0; k0 < CC; k0 += 4) {
    v2f a, b;
    a.x = F[(k0 + khalf + 0) * HW + colA];
    a.y = F[(k0 + khalf + 1) * HW + colA];
    b.x = F[(k0 + khalf + 0) * HW + colB];
    b.y = F[(k0 + khalf + 1) * HW + colB];
    acc = __builtin_amdgcn_wmma_f32_16x16x4_f32(false, a, false, b,
                                                (short)0, acc, false, false);
  }
  const int M0 = (lane >> 4) * 8;
  const int N  = lane & 15;
#pragma unroll
  for (int v = 0; v < 8; ++v) {
    G[(I * 16 + M0 + v) * HW + J * 16 + N] = acc[v];
  }
}

// ---------------------------------------------------------------------------
// make_grid + grid_sample_border position math (exactly replicates reference,
// including the double clip and the x*16/15 - 0.5 renormalization mismatch).
// ---------------------------------------------------------------------------
__device__ inline void bil_pos(float px, float py, int aidx, int pidx,
                               float length, int polar,
                               int& x0, int& x1, int& y0, int& y1,
                               float& wx, float& wy) {
  float ratio = (float)(pidx + 1) / (float)polar;
  float ang = (float)aidx * 15.0f * PI_F / 180.0f;
  float x = px + length * sinf(ang) * ratio;
  float y = py - length * cosf(ang) * ratio;
  x = fminf(fmaxf(x, 0.0f), 15.0f);
  y = fminf(fmaxf(y, 0.0f), 15.0f);
  float gx = x * (16.0f / 15.0f) - 0.5f;
  float gy = y * (16.0f / 15.0f) - 0.5f;
  gx = fminf(fmaxf(gx, 0.0f), 15.0f);
  gy = fminf(fmaxf(gy, 0.0f), 15.0f);
  float fx = floorf(gx), fy = floorf(gy);
  x0 = (int)fx; y0 = (int)fy;
  x1 = min(x0 + 1, 15); y1 = min(y0 + 1, 15);
  wx = gx - fx; wy = gy - fy;
}

// ---------------------------------------------------------------------------
// Kernel B: ss[g,hw] via Gram gathers; one block per pixel, one thread per g.
// num   = sum_k w_k G[idx_k, hw]
// |p|^2 = sum_{j,k} w_j w_k G[idx_j, idx_k]
// |f|   = sqrt(G[hw,hw]);  then relu + L2-normalize over g (96).
// ---------------------------------------------------------------------------
__global__ __launch_bounds__(96) void ss_kernel(const float* __restrict__ G,
                                                float* __restrict__ ss) {
  const int hw = blockIdx.x;
  const int px = hw & 15, py = hw >> 4;
  const int g = threadIdx.x;          // 0..95, g = a*4 + c
  const int a = g >> 2, c = g & 3;

  int x0, x1, y0, y1; float wx, wy;
  bil_pos((float)px, (float)py, a, c, 5.0f, DSC_POLAR, x0, x1, y0, y1, wx, wy);
  int   idx[4] = { y0 * 16 + x0, y0 * 16 + x1, y1 * 16 + x0, y1 * 16 + x1 };
  float w4[4]  = { (1.f - wx) * (1.f - wy), wx * (1.f - wy),
                   (1.f - wx) * wy,         wx * wy };

  float num = 0.f, den2 = 0.f;
#pragma unroll
  for (int k = 0; k < 4; ++k) num += w4[k] * G[idx[k] * HW + hw];
#pragma unroll
  for (int j = 0; j < 4; ++j)
#pragma unroll
    for (int k = 0; k < 4; ++k) den2 += w4[j] * w4[k] * G[idx[j] * HW + idx[k]];

  float pn = fmaxf(sqrtf(den2), EPSV);
  float fn = fmaxf(sqrtf(G[hw * HW + hw]), EPSV);
  float v  = fmaxf(num / (pn * fn), 0.0f);

  __shared__ float red[96];
  red[g] = v * v;
  __syncthreads();
  if (g == 0) {
    float s = 0.f;
    for (int i = 0; i < 96; ++i) s += red[i];
    red[0] = fmaxf(sqrtf(s), EPSV);
  }
  __syncthreads();
  ss[g * HW + hw] = v / red[0];
}

// ---------------------------------------------------------------------------
// Kernel C: sym_act via FIXED_KERNEL permutation sparsity.
// sigma_n(q) = (23 - q - n) mod 24;  w[n,q] = learnable[n, q, sigma_n(q)]
// sym_act[n] = sum_{o,q} w[n,o] w[n,q] sum_{c,d}
//              sym[o,c,q,d] * sym[sigma(o),c,sigma(q),d]
// One block per pixel; sym (96x72) staged in LDS. Then max(softmax) + concat.
// ---------------------------------------------------------------------------
__global__ __launch_bounds__(256) void symact_kernel(const float* __restrict__ ss,
                                                     const float* __restrict__ Lk,
                                                     float* __restrict__ out2) {
  const int hw = blockIdx.x;
  const int px = hw & 15, py = hw >> 4;
  const int tid = threadIdx.x;

  __shared__ float symL[96 * 72];
  __shared__ int   sidx[72][4];
  __shared__ float swt[72][4];
  __shared__ float part[576];
  __shared__ float sa[24];

  if (tid < 72) {                       // ray index = q*3 + d
    int q = tid / 3, d = tid - q * 3;
    int x0, x1, y0, y1; float wx, wy;
    bil_pos((float)px, (float)py, q, d, 8.0f, SYM_POLAR, x0, x1, y0, y1, wx, wy);
    sidx[tid][0] = y0 * 16 + x0; sidx[tid][1] = y0 * 16 + x1;
    sidx[tid][2] = y1 * 16 + x0; sidx[tid][3] = y1 * 16 + x1;
    swt[tid][0] = (1.f - wx) * (1.f - wy); swt[tid][1] = wx * (1.f - wy);
    swt[tid][2] = (1.f - wx) * wy;         swt[tid][3] = wx * wy;
  }
  __syncthreads();

  for (int e = tid; e < 96 * 72; e += 256) {
    int ch = e / 72, s = e - ch * 72;
    float acc = 0.f;
#pragma unroll
    for (int k = 0; k < 4; ++k) acc += swt[s][k] * ss[ch * HW + sidx[s][k]];
    symL[e] = acc;
  }
  __syncthreads();

  for (int p = tid; p < 576; p += 256) {
    int n = p / 24, o = p - n * 24;
    int so = (47 - o - n) % 24;
    float wno = Lk[(n * 24 + o) * 24 + so];
    float acc = 0.f;
    for (int q = 0; q < 24; ++q) {
      int sq = (47 - q - n) % 24;
      float wq = Lk[(n * 24 + q) * 24 + sq];
      float s = 0.f;
#pragma unroll
      for (int c = 0; c < 4; ++c)
#pragma unroll
        for (int d = 0; d < 3; ++d)
          s += symL[(o * 4 + c) * 72 + q * 3 + d] *
               symL[(so * 4 + c) * 72 + sq * 3 + d];
      acc += wq * s;
    }
    part[p] = wno * acc;
  }
  __syncthreads();

  if (tid < 24) {
    float s = 0.f;
    for (int o = 0; o < 24; ++o) s += part[tid * 24 + o];
    sa[tid] = s;
    out2[(1 + tid) * HW + hw] = s;      // channels 1..24 = sym_act
  }
  __syncthreads();
  if (tid == 0) {
    float m = sa[0];
    for (int i = 1; i < 24; ++i) m = fmaxf(m, sa[i]);
    float sum = 0.f;
    for (int i = 0; i < 24; ++i) sum += expf(sa[i] - m);
    out2[hw] = 1.0f / sum;              // channel 0 = max(softmax) = 1/sum
  }
}

extern "C" void kernel_launch(void* const* d_in, const int* in_sizes, int n_in,
                              void* d_out, int out_size, void* d_ws, size_t ws_size,
                              hipStream_t stream) {
  (void)in_sizes; (void)n_in; (void)out_size; (void)ws_size;
  const float* feature = (const float*)d_in[0];   // (1,512,16,16) f32
  const float* Lk      = (const float*)d_in[1];   // (24,24,24) f32
  float* out = (float*)d_out;                     // ss (96*256) ++ sym_act (25*256)
  float* G   = (float*)d_ws;                      // 256*256 f32 Gram scratch

  gram_kernel<<<32, 256, 0, stream>>>(feature, G);
  ss_kernel<<<HW, 96, 0, stream>>>(G, out);
  symact_kernel<<<HW, 256, 0, stream>>>(out, Lk, out + 96 * HW);
}